// JointModel_60988535603969
// MI455X (gfx1250) — compile-verified
//
#include <hip/hip_runtime.h>
#include <hip/hip_bf16.h>
#include <stdint.h>

typedef float v2f __attribute__((ext_vector_type(2)));
typedef float v8f __attribute__((ext_vector_type(8)));
typedef unsigned int v4u __attribute__((ext_vector_type(4)));
typedef int v4i __attribute__((ext_vector_type(4)));
typedef int v8i __attribute__((ext_vector_type(8)));

// ---------------------------------------------------------------------------
// WMMA f32 GEMM: C[M,128] = A[M,K] @ B[K,128], row-major, M % 16 == 0.
// Per block: TDM stages the 16xK A-tile into LDS once (pad +2 dwords/row for
// alignment + bank-conflict-free reads), then 8 waves each own a 16x16 output
// tile (8 waves = all 128 columns), accumulating with V_WMMA_F32_16X16X4_F32.
// A-frag (16x4 f32): lanes 0-15 hold K=k,k+1 for M=lane; lanes 16-31 K=k+2,k+3.
// B-frag (4x16 f32): mirrored (lane%16 = N column, half selects K pair).
// C/D  (16x16 f32): VGPR r, lane<16 -> row r, lane>=16 -> row 8+r.
// ---------------------------------------------------------------------------
template <int K>
__global__ __launch_bounds__(256) void gemm_f32_wmma(const float* __restrict__ A,
                                                     const float* __restrict__ B,
                                                     float* __restrict__ C) {
    constexpr int LROW = K + 2;  // padded LDS row stride in dwords
    __shared__ float As[16 * LROW];

    const int wave = threadIdx.x >> 5;
    const int lane = threadIdx.x & 31;
    const int half = lane >> 4;  // 0: K pair {k,k+1}, 1: K pair {k+2,k+3}
    const int l    = lane & 15;
    const int r0   = blockIdx.x * 16;
    const int c0   = wave * 16;

#if __has_builtin(__builtin_amdgcn_tensor_load_to_lds)
    if (wave == 0) {
        // ---- Tensor Data Mover: A[r0:r0+16, 0:K] -> LDS (row pad = 2 dwords)
        unsigned long long ga = (unsigned long long)(uintptr_t)(A + (size_t)r0 * K);
        unsigned lds = (unsigned)(uintptr_t)(void*)As;  // LDS byte offset (addr[31:0])
        constexpr unsigned padi = (K == 256) ? 7u : 6u; // pad interval: K dwords
        v4u g0;
        g0.x = 1u;                                           // count=1 (valid D#)
        g0.y = lds;                                          // lds_addr
        g0.z = (unsigned)ga;                                 // global_addr[31:0]
        g0.w = (unsigned)((ga >> 32) & 0x01FFFFFFull) | (2u << 30); // addr[56:32], type=2
        v8i g1;
        g1[0] = (int)((2u << 16)        // data_size = 4 bytes
                      | (1u << 20)      // pad_enable
                      | (padi << 22)    // pad_interval = K dwords
                      | (1u << 25));    // pad_amount  = 2 dwords
        g1[1] = (int)((unsigned)(K & 0xFFFF) << 16);  // tensor_dim0 = K (low 16)
        g1[2] = (int)(16u << 16);                     // tensor_dim0 hi=0 | tensor_dim1 = 16
        g1[3] = (int)((unsigned)(K & 0xFFFF) << 16);  // tensor_dim1 hi=0 | tile_dim0 = K
        g1[4] = 16;                                   // tile_dim1 = 16, tile_dim2 = 0
        g1[5] = K;                                    // tensor_dim0_stride = K (low 32)
        g1[6] = 0;                                    // stride0 hi | stride1 lo
        g1[7] = 0;                                    // stride1 hi
        v4i gz = {0, 0, 0, 0};
#if __clang_major__ >= 23
        v8i gz8 = {0, 0, 0, 0, 0, 0, 0, 0};
        __builtin_amdgcn_tensor_load_to_lds(g0, g1, gz, gz, gz8, 0);
#else
        __builtin_amdgcn_tensor_load_to_lds(g0, g1, gz, gz, 0);
#endif
        __builtin_amdgcn_s_wait_tensorcnt(0);
    }
#else
    // Fallback: cooperative vector staging of the A tile into LDS.
    for (int i = threadIdx.x; i < 16 * (K / 4); i += 256) {
        int r = i / (K / 4), kk = (i % (K / 4)) * 4;
        const float4 v = *(const float4*)(A + (size_t)(r0 + r) * K + kk);
        *(float4*)(&As[r * LROW + kk]) = v;
    }
#endif
    __syncthreads();

    v8f acc = {};
    const float* arow = As + l * LROW + half * 2;            // LDS, 8B aligned
    const float* bcol = B + (size_t)(half * 2) * 128 + c0 + l;

    for (int k = 0; k < K; k += 4) {
        v2f a, b;
        a.x = arow[k];
        a.y = arow[k + 1];
        b.x = bcol[0];
        b.y = bcol[128];
        acc = __builtin_amdgcn_wmma_f32_16x16x4_f32(
            /*neg_a=*/false, a, /*neg_b=*/false, b,
            /*c_mod=*/(short)0, acc, /*reuse_a=*/false, /*reuse_b=*/false);
        bcol += 4 * 128;
    }

    float* crow = C + (size_t)(r0 + half * 8) * 128 + c0 + l;
#pragma unroll
    for (int r = 0; r < 8; ++r) crow[(size_t)r * 128] = acc[r];
}

// ---------------------------------------------------------------------------
// deg[dst] += 1 per edge (self-loop +1 folded into dinv kernel)
// ---------------------------------------------------------------------------
__global__ __launch_bounds__(256) void deg_kernel(const long long* __restrict__ dst,
                                                  int E, float* __restrict__ deg) {
    int i = blockIdx.x * blockDim.x + threadIdx.x;
    if (i < E) atomicAdd(&deg[(int)dst[i]], 1.0f);
}

__global__ __launch_bounds__(256) void dinv_kernel(const float* __restrict__ deg,
                                                   float* __restrict__ dinv, int N) {
    int i = blockIdx.x * blockDim.x + threadIdx.x;
    if (i < N) dinv[i] = rsqrtf(deg[i] + 1.0f);  // +1 for self loop
}

// ---------------------------------------------------------------------------
// Edge scatter: acc[dst] += xw[src] * dinv[src]*dinv[dst].
// One wave per edge: 32 lanes x float4 = 128 features. Coalesced L2 atomics.
// ---------------------------------------------------------------------------
__global__ __launch_bounds__(256) void scatter_kernel(const long long* __restrict__ src,
                                                      const long long* __restrict__ dst,
                                                      int E,
                                                      const float* __restrict__ xw,
                                                      const float* __restrict__ dinv,
                                                      float* __restrict__ acc) {
    int tid = blockIdx.x * blockDim.x + threadIdx.x;
    int e = tid >> 5;
    if (e >= E) return;
    int f4 = (tid & 31) * 4;
    int s = (int)src[e];
    int d = (int)dst[e];
    float nrm = dinv[s] * dinv[d];
    const float4 v = *(const float4*)(xw + (size_t)s * 128 + f4);
    float* o = acc + (size_t)d * 128 + f4;
    atomicAdd(o + 0, v.x * nrm);
    atomicAdd(o + 1, v.y * nrm);
    atomicAdd(o + 2, v.z * nrm);
    atomicAdd(o + 3, v.w * nrm);
}

// ---------------------------------------------------------------------------
// out = acc + xw * dinv^2 (self loop) + bias, optional ReLU. N*128 threads.
// ---------------------------------------------------------------------------
__global__ __launch_bounds__(256) void finalize_kernel(const float* __restrict__ acc,
                                                       const float* __restrict__ xw,
                                                       const float* __restrict__ dinv,
                                                       const float* __restrict__ bias,
                                                       float* __restrict__ out,
                                                       int N, int do_relu) {
    int tid = blockIdx.x * blockDim.x + threadIdx.x;
    if (tid >= N * 128) return;
    int node = tid >> 7;
    int f = tid & 127;
    float di = dinv[node];
    float v = acc[tid] + xw[tid] * di * di + bias[f];
    if (do_relu) v = fmaxf(v, 0.0f);
    out[tid] = v;
}

// ---------------------------------------------------------------------------
// Node head: logits = relu(h) @ Wcls[128,10] + bcls, then log_softmax.
// One wave (32 lanes) per node; lane-parallel partial dots + shfl reduction.
// ---------------------------------------------------------------------------
__global__ __launch_bounds__(256) void node_head_kernel(const float* __restrict__ h,
                                                        const float* __restrict__ Wcls,
                                                        const float* __restrict__ bcls,
                                                        float* __restrict__ out, int N) {
    int wid = (blockIdx.x * blockDim.x + threadIdx.x) >> 5;
    int lane = threadIdx.x & 31;
    if (wid >= N) return;

    int f0 = lane * 4;
    float4 hv = *(const float4*)(h + (size_t)wid * 128 + f0);
    hv.x = fmaxf(hv.x, 0.0f);
    hv.y = fmaxf(hv.y, 0.0f);
    hv.z = fmaxf(hv.z, 0.0f);
    hv.w = fmaxf(hv.w, 0.0f);

    float logits[10];
#pragma unroll
    for (int c = 0; c < 10; ++c) {
        const float* wc = Wcls + (size_t)f0 * 10 + c;
        float p = hv.x * wc[0] + hv.y * wc[10] + hv.z * wc[20] + hv.w * wc[30];
#pragma unroll
        for (int o = 16; o > 0; o >>= 1) p += __shfl_down(p, o, 32);
        logits[c] = p;  // valid in lane 0
    }

    if (lane == 0) {
        float m = -3.402823e38f;
#pragma unroll
        for (int c = 0; c < 10; ++c) {
            logits[c] += bcls[c];
            m = fmaxf(m, logits[c]);
        }
        float s = 0.0f;
#pragma unroll
        for (int c = 0; c < 10; ++c) s += expf(logits[c] - m);
        float lse = m + logf(s);
        float* o = out + (size_t)wid * 10;
#pragma unroll
        for (int c = 0; c < 10; ++c) o[c] = logits[c] - lse;
    }
}

// ---------------------------------------------------------------------------
// Link head: out[e] = dot(h[s], Wl[0:128]) + dot(h[d], Wl[128:256]) + blink.
// One wave per edge; coalesced 128-float row loads for both endpoints.
// ---------------------------------------------------------------------------
__global__ __launch_bounds__(256) void link_head_kernel(const float* __restrict__ h,
                                                        const long long* __restrict__ ei,
                                                        int E,
                                                        const float* __restrict__ Wl,
                                                        const float* __restrict__ blink,
                                                        float* __restrict__ out) {
    int wid = (blockIdx.x * blockDim.x + threadIdx.x) >> 5;
    int lane = threadIdx.x & 31;
    if (wid >= E) return;

    int s = (int)ei[wid];
    int d = (int)ei[(size_t)E + wid];
    int f0 = lane * 4;

    float4 hs = *(const float4*)(h + (size_t)s * 128 + f0);
    float4 hd = *(const float4*)(h + (size_t)d * 128 + f0);
    float4 ws = *(const float4*)(Wl + f0);
    float4 wd = *(const float4*)(Wl + 128 + f0);

    float p = hs.x * ws.x + hs.y * ws.y + hs.z * ws.z + hs.w * ws.w +
              hd.x * wd.x + hd.y * wd.y + hd.z * wd.z + hd.w * wd.w;
#pragma unroll
    for (int o = 16; o > 0; o >>= 1) p += __shfl_down(p, o, 32);

    if (lane == 0) out[wid] = p + blink[0];
}

// ---------------------------------------------------------------------------
extern "C" void kernel_launch(void* const* d_in, const int* in_sizes, int n_in,
                              void* d_out, int out_size, void* d_ws, size_t ws_size,
                              hipStream_t stream) {
    const float*     x     = (const float*)d_in[0];
    const long long* ei    = (const long long*)d_in[1];
    const long long* eip   = (const long long*)d_in[2];
    const long long* ein   = (const long long*)d_in[3];
    const float*     W1    = (const float*)d_in[4];
    const float*     b1    = (const float*)d_in[5];
    const float*     W2    = (const float*)d_in[6];
    const float*     b2    = (const float*)d_in[7];
    const float*     Wcls  = (const float*)d_in[8];
    const float*     bcls  = (const float*)d_in[9];
    const float*     Wlink = (const float*)d_in[10];
    const float*     blink = (const float*)d_in[11];

    const int N  = in_sizes[0] / 256;  // 100000
    const int E  = in_sizes[1] / 2;    // 1600000
    const int Ep = in_sizes[2] / 2;    // 200000
    const int En = in_sizes[3] / 2;    // 200000

    // Workspace layout (floats): xw | acc | h | deg | dinv  (~155 MB)
    float* xw   = (float*)d_ws;
    float* acc  = xw  + (size_t)N * 128;
    float* h    = acc + (size_t)N * 128;
    float* deg  = h   + (size_t)N * 128;
    float* dinv = deg + N;

    const long long* src = ei;
    const long long* dst = ei + E;

    // Degree + normalization (shared by both layers)
    hipMemsetAsync(deg, 0, sizeof(float) * (size_t)N, stream);
    deg_kernel<<<(E + 255) / 256, 256, 0, stream>>>(dst, E, deg);
    dinv_kernel<<<(N + 255) / 256, 256, 0, stream>>>(deg, dinv, N);

    // ---- Layer 1: h = relu( scatter(norm * x@W1) + selfloop + b1 ) ----
    gemm_f32_wmma<256><<<N / 16, 256, 0, stream>>>(x, W1, xw);
    hipMemsetAsync(acc, 0, sizeof(float) * (size_t)N * 128, stream);
    scatter_kernel<<<(E * 32 + 255) / 256, 256, 0, stream>>>(src, dst, E, xw, dinv, acc);
    finalize_kernel<<<(N * 128 + 255) / 256, 256, 0, stream>>>(acc, xw, dinv, b1, h, N, 1);

    // ---- Layer 2: h = scatter(norm * h@W2) + selfloop + b2 ----
    gemm_f32_wmma<128><<<N / 16, 256, 0, stream>>>(h, W2, xw);
    hipMemsetAsync(acc, 0, sizeof(float) * (size_t)N * 128, stream);
    scatter_kernel<<<(E * 32 + 255) / 256, 256, 0, stream>>>(src, dst, E, xw, dinv, acc);
    finalize_kernel<<<(N * 128 + 255) / 256, 256, 0, stream>>>(acc, xw, dinv, b2, h, N, 0);

    // ---- Heads ----
    float* out = (float*)d_out;
    node_head_kernel<<<(N * 32 + 255) / 256, 256, 0, stream>>>(h, Wcls, bcls, out, N);
    link_head_kernel<<<(Ep * 32 + 255) / 256, 256, 0, stream>>>(h, eip, Ep, Wlink, blink,
                                                                out + (size_t)N * 10);
    link_head_kernel<<<(En * 32 + 255) / 256, 256, 0, stream>>>(h, ein, En, Wlink, blink,
                                                                out + (size_t)N * 10 + Ep);
}